// InvariantMessagePasser_49984829391467
// MI455X (gfx1250) — compile-verified
//
#include <hip/hip_runtime.h>

// ---------------------------------------------------------------------------
// InvariantMessagePasser for MI455X (gfx1250, wave32).
//
// Fast path: counting-sort edges by center (packed (edge,neighbor) pairs),
// then one wave per atom computes
//   dens_l^T(k,m) = sum_edges w_l(e,k) * sh_l(e,m),  w_l = rb_l * emb[nbr]
// as V_WMMA_F32_16X16X4_F32 ops (M = k tile of 16, N = m, K = 4 edges/chunk).
// Accumulators stay in WMMA C tiles; output written once (no f32 atomics).
// All hot-loop loads are written as UNCONDITIONAL loads (padding slots hold
// valid ids; tail indices are clamped) followed by value selects, so the
// compiler emits plain global_load + v_cndmask instead of exec diamonds.
// Fallback path (small ws): coalesced-lane atomic scatter.
// ---------------------------------------------------------------------------

#ifndef MP_SCALING
#define MP_SCALING 0.1f
#endif

typedef __attribute__((ext_vector_type(2))) float v2f;
typedef __attribute__((ext_vector_type(8))) float v8f;

// ===========================================================================
// Sort pipeline kernels
// ===========================================================================
// Zero histogram + initialize order2 padding slots with safe (0,0) ids so the
// gather loop can load them unconditionally.
__global__ __launch_bounds__(256) void imp_wszero(int* __restrict__ cnt, int n,
                                                  int2* __restrict__ order2,
                                                  int n_edges) {
  int i = blockIdx.x * 256 + threadIdx.x;
  if (i < n) cnt[i] = 0;
  if (blockIdx.x == 0 && threadIdx.x < 8)
    order2[n_edges + threadIdx.x] = make_int2(0, 0);
}

__global__ __launch_bounds__(256) void imp_hist(const int* __restrict__ centers,
                                                int* __restrict__ cnt,
                                                int n_edges) {
  int e = blockIdx.x * 256 + threadIdx.x;
  if (e < n_edges) atomicAdd(&cnt[centers[e]], 1);
}

// Per-256-block exclusive scan; block totals to bsum.
__global__ __launch_bounds__(256) void imp_scan1(const int* __restrict__ cnt,
                                                 int* __restrict__ offs,
                                                 int* __restrict__ bsum,
                                                 int n) {
  __shared__ int tmp[256];
  const int i = blockIdx.x * 256 + threadIdx.x;
  const int v = (i < n) ? cnt[i] : 0;
  tmp[threadIdx.x] = v;
  __syncthreads();
  int acc = v;
#pragma unroll
  for (int off = 1; off < 256; off <<= 1) {
    int t = (threadIdx.x >= off) ? tmp[threadIdx.x - off] : 0;
    __syncthreads();
    acc += t;
    tmp[threadIdx.x] = acc;
    __syncthreads();
  }
  if (i < n) offs[i] = acc - v;                    // local exclusive
  if (threadIdx.x == 255) bsum[blockIdx.x] = acc;  // block total
}

// Exclusive scan of up to 1024 block totals, in place.
__global__ __launch_bounds__(1024) void imp_scan2(int* __restrict__ bsum,
                                                  int nb) {
  __shared__ int tmp[1024];
  const int v = (threadIdx.x < nb) ? bsum[threadIdx.x] : 0;
  tmp[threadIdx.x] = v;
  __syncthreads();
  int acc = v;
#pragma unroll
  for (int off = 1; off < 1024; off <<= 1) {
    int t = (threadIdx.x >= off) ? tmp[threadIdx.x - off] : 0;
    __syncthreads();
    acc += t;
    tmp[threadIdx.x] = acc;
    __syncthreads();
  }
  if (threadIdx.x < nb) bsum[threadIdx.x] = acc - v;
}

// Add block offsets; produce global exclusive offsets + scatter cursor.
__global__ __launch_bounds__(256) void imp_scan3(int* __restrict__ offs,
                                                 int* __restrict__ cursor,
                                                 const int* __restrict__ bsum,
                                                 int n, int n_edges) {
  int i = blockIdx.x * 256 + threadIdx.x;
  if (i < n) {
    int o = offs[i] + bsum[i >> 8];
    offs[i] = o;
    cursor[i] = o;
  }
  if (i == 0) offs[n] = n_edges;
}

// Scatter packed (edge, neighbor) pairs -> one s_load_b256 per chunk later,
// no dependent neighbor gather in the hot loop.
__global__ __launch_bounds__(256) void imp_place(
    const int* __restrict__ centers, const int* __restrict__ neighbors,
    int* __restrict__ cursor, int2* __restrict__ order2, int n_edges) {
  int e = blockIdx.x * 256 + threadIdx.x;
  if (e < n_edges) {
    int pos = atomicAdd(&cursor[centers[e]], 1);
    order2[pos] = make_int2(e, neighbors[e]);
  }
}

// ===========================================================================
// Per-atom WMMA gather kernel
// ===========================================================================
// One V_WMMA_F32_16X16X4_F32 per (l, k-tile) per 4-edge chunk:
//   A(16x4): M = k-within-tile, K = edge slot.  lane = M; VGPR = K&1; half = K>>1
//   B(4x16): K = edge slot, N = m.              lane = N; VGPR = K&1; half = K>>1
//   C(16x16): VGPR q, lane half h -> k = q + 8h (+16*tile), m = lane&15
//
// Invalid (padding) slots are zeroed purely through the emb factors: a zero
// emb zeroes the A column, so the whole K-column contribution vanishes and B
// needs no validity mask. All loads below are unconditional (clamped
// addresses); masks are applied to the loaded VALUES (v_cndmask), never to
// the loads, so no exec-mask diamonds are generated.
__device__ __forceinline__ void mma_l(const float* __restrict__ sh,
                                      const float* __restrict__ rb, int TL,
                                      int KL, int eA, int eB, int kk,
                                      float embA0, float embA1, float embB0,
                                      float embB1, v8f& c0, v8f& c1) {
  const int k2 = (kk + 16 < KL) ? kk + 16 : KL - 1;  // clamped, in-bounds
  const bool hi = (kk + 16 < KL);                    // lane-constant mask
  const int mk = (kk < TL) ? kk : TL - 1;            // clamped, in-bounds
  const bool bm = (kk < TL);                         // lane-constant mask

  // Unconditional loads.
  const float rA0 = rb[eA * KL + kk];
  const float rB0 = rb[eB * KL + kk];
  const float rA1 = rb[eA * KL + k2];
  const float rB1 = rb[eB * KL + k2];
  const float sA = sh[eA * TL + mk];
  const float sB = sh[eB * TL + mk];

  v2f a0, a1, b;
  a0.x = rA0 * embA0;
  a0.y = rB0 * embB0;
  a1.x = hi ? rA1 * embA1 : 0.f;
  a1.y = hi ? rB1 * embB1 : 0.f;
  b.x = bm ? sA : 0.f;
  b.y = bm ? sB : 0.f;
  c0 = __builtin_amdgcn_wmma_f32_16x16x4_f32(false, a0, false, b, (short)0, c0,
                                             false, false);
  c1 = __builtin_amdgcn_wmma_f32_16x16x4_f32(false, a1, false, b, (short)0, c1,
                                             false, false);
}

// Scale + dump one (l) pair of C tiles into the per-wave LDS buffer at rows
// [s, s+TL), columns = k.
__device__ __forceinline__ void dump_l(float* __restrict__ rows /*[16][32]*/,
                                       int s, int TL, int kk, int h,
                                       const v8f& c0, const v8f& c1) {
  if (kk < TL) {
    float* r = rows + (s + kk) * 32;
#pragma unroll
    for (int q = 0; q < 8; ++q) r[q + 8 * h] = MP_SCALING * c0[q];
#pragma unroll
    for (int q = 0; q < 8; ++q) r[16 + q + 8 * h] = MP_SCALING * c1[q];
  }
}

__global__ __launch_bounds__(256) void imp_gather(
    const float* __restrict__ sh0, const float* __restrict__ sh1,
    const float* __restrict__ sh2, const float* __restrict__ sh3,
    const float* __restrict__ rb0, const float* __restrict__ rb1,
    const float* __restrict__ rb2, const float* __restrict__ rb3,
    const float* __restrict__ emb, const int* __restrict__ offs,
    const int2* __restrict__ order2, float* __restrict__ out, int natoms,
    int o1, int o2, int o3) {
  __shared__ float lds[8][16][32];  // 8 waves/block, 2KB per wave
  const int lane = threadIdx.x & 31;
  const int wv = threadIdx.x >> 5;
  int a = blockIdx.x * 8 + wv;
  if (a >= natoms) return;
  a = __builtin_amdgcn_readfirstlane(a);  // whole pipeline wave-uniform

  const int h = lane >> 4;   // lane half
  const int kk = lane & 15;  // 0..15

  const int beg = offs[a];
  const int end = offs[a + 1];

  v8f c00 = {}, c01 = {}, c10 = {}, c11 = {};
  v8f c20 = {}, c21 = {}, c30 = {}, c31 = {};

  for (int p = beg; p < end; p += 4) {
    __builtin_prefetch(order2 + p + 4, 0, 3);  // gfx1250 global_prefetch_b8
    // 4 edge slots: one merged s_load_b256 (order2 is padded with valid (0,0)
    // entries, so p+3 is always loadable) — branch-free chunk setup.
    const int2 q0 = order2[p];
    const int2 q1 = order2[p + 1];
    const int2 q2 = order2[p + 2];
    const int2 q3 = order2[p + 3];
    const bool v1 = (p + 1 < end), v2 = (p + 2 < end), v3 = (p + 3 < end);

    // Lane-half -> edge-slot mapping dictated by the A/B VGPR layouts.
    const int eA = h ? q2.x : q0.x;
    const int nA = h ? q2.y : q0.y;
    const int eB = h ? q3.x : q1.x;
    const int nB = h ? q3.y : q1.y;
    const bool vA = h ? v2 : true;
    const bool vB = h ? v3 : v1;

    // Gathered neighbor embedding, k = kk and kk+16. Loads are unconditional
    // (ids always valid); invalid slots are zeroed by a VALUE select, which
    // zeroes the whole A column below. Do not fold the condition into the
    // load expression — that would re-create predicated-load diamonds.
    const float tA0 = emb[nA * 32 + kk];
    const float tA1 = emb[nA * 32 + kk + 16];
    const float tB0 = emb[nB * 32 + kk];
    const float tB1 = emb[nB * 32 + kk + 16];
    const float embA0 = vA ? tA0 : 0.f;
    const float embA1 = vA ? tA1 : 0.f;
    const float embB0 = vB ? tB0 : 0.f;
    const float embB1 = vB ? tB1 : 0.f;

    mma_l(sh0, rb0, 1, 32, eA, eB, kk, embA0, embA1, embB0, embB1, c00, c01);
    mma_l(sh1, rb1, 3, 28, eA, eB, kk, embA0, embA1, embB0, embB1, c10, c11);
    mma_l(sh2, rb2, 5, 24, eA, eB, kk, embA0, embA1, embB0, embB1, c20, c21);
    mma_l(sh3, rb3, 7, 20, eA, eB, kk, embA0, embA1, embB0, embB1, c30, c31);
  }

  // Transpose C tiles (k-major) to m-major rows via this wave's LDS buffer.
  // Rows: l0 -> 0, l1 -> 1..3, l2 -> 4..8, l3 -> 9..15. Same-wave DS ops are
  // in order, so no barrier is required (buffer is private to the wave).
  float* rows = &lds[wv][0][0];
  dump_l(rows, 0, 1, kk, h, c00, c01);
  dump_l(rows, 1, 3, kk, h, c10, c11);
  dump_l(rows, 4, 5, kk, h, c20, c21);
  dump_l(rows, 9, 7, kk, h, c30, c31);

  // Coalesced output rows (center index == atom index).
  out[a * 32 + lane] = lds[wv][0][lane];  // l=0, k_l=32
  if (lane < 28) {
    float* b = out + o1 + a * 84;  // 3*28
#pragma unroll
    for (int m = 0; m < 3; ++m) b[m * 28 + lane] = lds[wv][1 + m][lane];
  }
  if (lane < 24) {
    float* b = out + o2 + a * 120;  // 5*24
#pragma unroll
    for (int m = 0; m < 5; ++m) b[m * 24 + lane] = lds[wv][4 + m][lane];
  }
  if (lane < 20) {
    float* b = out + o3 + a * 140;  // 7*20
#pragma unroll
    for (int m = 0; m < 7; ++m) b[m * 20 + lane] = lds[wv][9 + m][lane];
  }
}

// ===========================================================================
// Fallback path (workspace too small): zero + coalesced-lane atomic scatter.
// ===========================================================================
__global__ __launch_bounds__(256) void imp_zero(float4* __restrict__ out4,
                                                float* __restrict__ out,
                                                int n4, int ntail) {
  int i = blockIdx.x * 256 + threadIdx.x;
  if (i < n4) out4[i] = make_float4(0.f, 0.f, 0.f, 0.f);
  if (i == 0)
    for (int t = 0; t < ntail; ++t) out[4 * n4 + t] = 0.f;
}

__global__ __launch_bounds__(256) void imp_scatter(
    const float* __restrict__ sh0, const float* __restrict__ sh1,
    const float* __restrict__ sh2, const float* __restrict__ sh3,
    const float* __restrict__ rb0, const float* __restrict__ rb1,
    const float* __restrict__ rb2, const float* __restrict__ rb3,
    const float* __restrict__ emb, const int* __restrict__ centers,
    const int* __restrict__ neighbors, float* __restrict__ out, int n_edges,
    int o1, int o2, int o3) {
  const int lane = threadIdx.x & 31;
  int e = blockIdx.x * 8 + (threadIdx.x >> 5);
  if (e >= n_edges) return;
  e = __builtin_amdgcn_readfirstlane(e);
  const int c = centers[e];
  const int n = neighbors[e];
  const float embv = emb[n * 32 + lane];
  {
    const float w = rb0[e * 32 + lane] * embv;
    atomicAdd(out + c * 32 + lane, (MP_SCALING * sh0[e]) * w);
  }
  if (lane < 28) {
    const float w = rb1[e * 28 + lane] * embv;
    float* b = out + o1 + c * 84 + lane;
#pragma unroll
    for (int m = 0; m < 3; ++m)
      atomicAdd(b + m * 28, (MP_SCALING * sh1[e * 3 + m]) * w);
  }
  if (lane < 24) {
    const float w = rb2[e * 24 + lane] * embv;
    float* b = out + o2 + c * 120 + lane;
#pragma unroll
    for (int m = 0; m < 5; ++m)
      atomicAdd(b + m * 24, (MP_SCALING * sh2[e * 5 + m]) * w);
  }
  if (lane < 20) {
    const float w = rb3[e * 20 + lane] * embv;
    float* b = out + o3 + c * 140 + lane;
#pragma unroll
    for (int m = 0; m < 7; ++m)
      atomicAdd(b + m * 20, (MP_SCALING * sh3[e * 7 + m]) * w);
  }
}

// ===========================================================================
// Launch
// ===========================================================================
extern "C" void kernel_launch(void* const* d_in, const int* in_sizes, int n_in,
                              void* d_out, int out_size, void* d_ws,
                              size_t ws_size, hipStream_t stream) {
  const float* sh0 = (const float*)d_in[0];
  const float* sh1 = (const float*)d_in[1];
  const float* sh2 = (const float*)d_in[2];
  const float* sh3 = (const float*)d_in[3];
  const float* rb0 = (const float*)d_in[4];
  const float* rb1 = (const float*)d_in[5];
  const float* rb2 = (const float*)d_in[6];
  const float* rb3 = (const float*)d_in[7];
  const float* emb = (const float*)d_in[8];
  const int* centers = (const int*)d_in[9];
  const int* neighbors = (const int*)d_in[10];
  float* out = (float*)d_out;

  const int n_edges = in_sizes[0];    // sh0 is [n_edges, 1, 1]
  const int natoms = out_size / 376;  // 32 + 84 + 120 + 140
  const int o1 = natoms * 32;
  const int o2 = o1 + natoms * 84;
  const int o3 = o2 + natoms * 120;

  // Workspace layout: cnt/cursor | offs | bsum(1025, keeps order2 8B-aligned)
  //                   | order2 (n_edges + 8 padding entries of int2)
  const int nb = (natoms + 255) / 256;
  int* cnt = (int*)d_ws;
  int* offs = cnt + natoms;
  int* bsum = offs + natoms + 1;
  int2* order2 = (int2*)(bsum + 1025);
  const size_t need =
      ((size_t)natoms * 2 + 1026) * 4 + ((size_t)n_edges + 8) * 8;

  if (ws_size >= need && nb <= 1024) {
    const int gbE = (n_edges + 255) / 256;
    const int gbA = (natoms + 255) / 256;
    imp_wszero<<<gbA, 256, 0, stream>>>(cnt, natoms, order2, n_edges);
    imp_hist<<<gbE, 256, 0, stream>>>(centers, cnt, n_edges);
    imp_scan1<<<nb, 256, 0, stream>>>(cnt, offs, bsum, natoms);
    imp_scan2<<<1, 1024, 0, stream>>>(bsum, nb);
    imp_scan3<<<gbA, 256, 0, stream>>>(offs, cnt /*cursor*/, bsum, natoms,
                                       n_edges);
    imp_place<<<gbE, 256, 0, stream>>>(centers, neighbors, cnt, order2,
                                       n_edges);
    imp_gather<<<(natoms + 7) / 8, 256, 0, stream>>>(
        sh0, sh1, sh2, sh3, rb0, rb1, rb2, rb3, emb, offs, order2, out, natoms,
        o1, o2, o3);
  } else {
    const int n4 = out_size >> 2;
    imp_zero<<<(n4 + 255) / 256, 256, 0, stream>>>((float4*)d_out, out, n4,
                                                   out_size & 3);
    imp_scatter<<<(n_edges + 7) / 8, 256, 0, stream>>>(
        sh0, sh1, sh2, sh3, rb0, rb1, rb2, rb3, emb, centers, neighbors, out,
        n_edges, o1, o2, o3);
  }
}